// SenarioEncoder_25271587569804
// MI455X (gfx1250) — compile-verified
//
#include <hip/hip_runtime.h>
#include <math.h>

#define T_ 50
#define A_ 64
#define P_ 512
#define HID_ 128
#define NSRC_ (T_*A_)   // 3200
#define NDST_ (T_*P_)   // 25600
#define PI_ 3.14159265358979323846f
#define TWO_PI_ 6.28318530717958647692f

typedef __attribute__((ext_vector_type(2))) float v2f;
typedef __attribute__((ext_vector_type(8))) float v8f;
typedef __attribute__((ext_vector_type(4))) int v4i;

#if defined(__has_builtin)
#if __has_builtin(__builtin_amdgcn_global_load_async_to_lds_b128) && \
    __has_builtin(__builtin_amdgcn_s_wait_asynccnt)
#define USE_ASYNC_G2L 1
#endif
#endif

#ifdef USE_ASYNC_G2L
typedef __attribute__((address_space(1))) v4i* gv4i_p;
typedef __attribute__((address_space(3))) v4i* lv4i_p;
#endif

// Per-lane 16B async global->LDS copy (ASYNCcnt path) with portable fallback.
static __device__ __forceinline__ void g2l_b128(float* lds, const float* g) {
#ifdef USE_ASYNC_G2L
  __builtin_amdgcn_global_load_async_to_lds_b128((gv4i_p)(float*)g, (lv4i_p)lds, 0, 0);
#else
  *(float4*)lds = *(const float4*)g;
#endif
}
static __device__ __forceinline__ void g2l_fence() {
#ifdef USE_ASYNC_G2L
  __builtin_amdgcn_s_wait_asynccnt(0);
#endif
}

static __device__ __forceinline__ v8f vzero8() {
  v8f z;
#pragma unroll
  for (int i = 0; i < 8; ++i) z[i] = 0.f;
  return z;
}

static __device__ __forceinline__ v8f wmma_f32(v2f a, v2f b, v8f c) {
  // D = A(16x4,f32) * B(4x16,f32) + C(16x16,f32); wave32, EXEC all-ones here.
  return __builtin_amdgcn_wmma_f32_16x16x4_f32(false, a, false, b, (short)0, c, false, false);
}

// A-fragment from LDS (A is 16 x K row-major, stride lda).
// ISA layout: lanes 0-15 hold M=lane, K=k..k+1 ; lanes 16-31 hold M=lane-16, K=k+2..k+3.
static __device__ __forceinline__ v2f lda_f(const float* sm, int lda, int k, int lane) {
  int m = lane & 15;
  int kk = k + ((lane >> 4) << 1);
  v2f a;
  a[0] = sm[m * lda + kk];
  a[1] = sm[m * lda + kk + 1];
  return a;
}

// B-fragment from global row-major B (K x N, stride ldb), col base n0; rows >= kmax read as 0.
static __device__ __forceinline__ v2f ldb_f(const float* B, int ldb, int k, int n0, int lane, int kmax) {
  int n = n0 + (lane & 15);
  int kk = k + ((lane >> 4) << 1);
  v2f b;
  b[0] = (kk     < kmax) ? B[(size_t)kk * ldb + n]       : 0.f;
  b[1] = (kk + 1 < kmax) ? B[(size_t)(kk + 1) * ldb + n] : 0.f;
  return b;
}

// C = A(16xK from LDS) * B(KxN global) accumulated into c0 (cols n0..n0+15) and c1 (n0+16..n0+31).
static __device__ __forceinline__ void gemm16(const float* smA, int lda, const float* B, int ldb,
                                              int K, int kmax, int n0, int lane, v8f& c0, v8f& c1) {
  for (int k = 0; k < K; k += 4) {
    v2f a  = lda_f(smA, lda, k, lane);
    v2f b0 = ldb_f(B, ldb, k, n0, lane, kmax);
    v2f b1 = ldb_f(B, ldb, k, n0 + 16, lane, kmax);
    c0 = wmma_f32(a, b0, c0);
    c1 = wmma_f32(a, b1, c1);
  }
}

// C/D layout: VGPR r -> row r (lanes 0-15) / row r+8 (lanes 16-31), col = n0 + (lane&15).
static __device__ __forceinline__ void store_c_lds(float* sm, int ld, int n0, int lane,
                                                   const v8f c0, const v8f c1) {
  int col = lane & 15;
  int rb = (lane >> 4) << 3;
#pragma unroll
  for (int r = 0; r < 8; ++r) {
    sm[(rb + r) * ld + n0 + col]      = c0[r];
    sm[(rb + r) * ld + n0 + 16 + col] = c1[r];
  }
}

static __device__ __forceinline__ float sigm(float x) { return 1.f / (1.f + expf(-x)); }

// Row-wise LayerNorm over n (=128) cols of a 16-row LDS tile; 128 threads; red = 256 floats scratch.
static __device__ void ln_rows(float* sm, int ld, int n, const float* g, const float* b,
                               float* red, int tid, bool relu) {
  int row = tid & 15, sub = tid >> 4;
  int per = n >> 3;
  int j0 = sub * per;
  float s = 0.f, s2 = 0.f;
  for (int j = 0; j < per; ++j) { float v = sm[row * ld + j0 + j]; s += v; s2 += v * v; }
  red[row * 8 + sub] = s;
  red[128 + row * 8 + sub] = s2;
  __syncthreads();
  float mu = 0.f, ms = 0.f;
  for (int i = 0; i < 8; ++i) { mu += red[row * 8 + i]; ms += red[128 + row * 8 + i]; }
  float inv_n = 1.f / (float)n;
  mu *= inv_n;
  ms = ms * inv_n - mu * mu;
  float inv = rsqrtf(ms + 1e-5f);
  for (int j = 0; j < per; ++j) {
    int c = j0 + j;
    float v = (sm[row * ld + c] - mu) * inv * g[c] + b[c];
    if (relu) v = fmaxf(v, 0.f);
    sm[row * ld + c] = v;
  }
  __syncthreads();
}

// ---------------------------------------------------------------------------
__global__ void k_init(unsigned* __restrict__ menc, float* __restrict__ den,
                       float* __restrict__ agg, float* __restrict__ hbuf,
                       const float* __restrict__ h0) {
  int i = blockIdx.x * blockDim.x + threadIdx.x;
  if (i < NDST_ * 8) { menc[i] = 0x007FFFFFu; /* enc(-inf) */ den[i] = 0.f; }
  if (i < NDST_ * HID_) agg[i] = 0.f;
  if (i < P_ * HID_) hbuf[i] = h0[i & 127];
}

// xs = LN(agent_tok) ; kbuf = xs@Wk ; vbuf = xs@Wv + bv   (rows: time-major agents)
__global__ __launch_bounds__(128) void k_src(const float* __restrict__ agent_enc,
                                             const float* __restrict__ g, const float* __restrict__ b,
                                             const float* __restrict__ Wk, const float* __restrict__ Wv,
                                             const float* __restrict__ bv,
                                             float* __restrict__ kbuf, float* __restrict__ vbuf) {
  __shared__ float sm[16 * 128];
  __shared__ float red[256];
  const int tid = threadIdx.x, lane = tid & 31, wave = tid >> 5;
  const int i0 = blockIdx.x * 16;
  const int n0 = wave * 32;
  if (tid == 0) { __builtin_prefetch(Wk, 0, 1); __builtin_prefetch(Wv, 0, 1); }
  for (int idx = tid; idx < 512; idx += 128) {
    int m = idx >> 5, c = (idx & 31) << 2;
    int row = i0 + m, a = row % A_, t = row / A_;
    g2l_b128(sm + m * 128 + c, agent_enc + ((size_t)a * T_ + t) * HID_ + c);
  }
  g2l_fence();
  __syncthreads();
  ln_rows(sm, 128, 128, g, b, red, tid, false);
  {
    v8f c0 = vzero8(), c1 = vzero8();
    gemm16(sm, 128, Wk, 128, 128, 128, n0, lane, c0, c1);
    int col = lane & 15, rb = (lane >> 4) << 3;
    for (int r = 0; r < 8; ++r) {
      kbuf[(size_t)(i0 + rb + r) * 128 + n0 + col]      = c0[r];
      kbuf[(size_t)(i0 + rb + r) * 128 + n0 + 16 + col] = c1[r];
    }
  }
  {
    v8f c0 = vzero8(), c1 = vzero8();
    gemm16(sm, 128, Wv, 128, 128, 128, n0, lane, c0, c1);
    int col = lane & 15, rb = (lane >> 4) << 3;
    for (int r = 0; r < 8; ++r) {
      vbuf[(size_t)(i0 + rb + r) * 128 + n0 + col]      = c0[r] + bv[n0 + col];
      vbuf[(size_t)(i0 + rb + r) * 128 + n0 + 16 + col] = c1[r] + bv[n0 + 16 + col];
    }
  }
}

// xd = LN(map_enc) ; qbuf = xd@Wq+bq ; sbuf = xd@Ws+bs ; xdln stored (only 512 rows needed)
__global__ __launch_bounds__(128) void k_dst(const float* __restrict__ map_enc,
                                             const float* __restrict__ g, const float* __restrict__ b,
                                             const float* __restrict__ Wq, const float* __restrict__ bq,
                                             const float* __restrict__ Ws, const float* __restrict__ bs,
                                             float* __restrict__ qbuf, float* __restrict__ sbuf,
                                             float* __restrict__ xdln) {
  __shared__ float sm[16 * 128];
  __shared__ float red[256];
  const int tid = threadIdx.x, lane = tid & 31, wave = tid >> 5;
  const int i0 = blockIdx.x * 16;
  const int n0 = wave * 32;
  if (tid == 0) { __builtin_prefetch(Wq, 0, 1); __builtin_prefetch(Ws, 0, 1); }
  for (int idx = tid; idx < 512; idx += 128)
    g2l_b128(sm + idx * 4, map_enc + (size_t)i0 * 128 + idx * 4);
  g2l_fence();
  __syncthreads();
  ln_rows(sm, 128, 128, g, b, red, tid, false);
  for (int idx = tid; idx < 2048; idx += 128) xdln[(size_t)i0 * 128 + idx] = sm[idx];
  {
    v8f c0 = vzero8(), c1 = vzero8();
    gemm16(sm, 128, Wq, 128, 128, 128, n0, lane, c0, c1);
    int col = lane & 15, rb = (lane >> 4) << 3;
    for (int r = 0; r < 8; ++r) {
      qbuf[(size_t)(i0 + rb + r) * 128 + n0 + col]      = c0[r] + bq[n0 + col];
      qbuf[(size_t)(i0 + rb + r) * 128 + n0 + 16 + col] = c1[r] + bq[n0 + 16 + col];
    }
  }
  {
    v8f c0 = vzero8(), c1 = vzero8();
    gemm16(sm, 128, Ws, 128, 128, 128, n0, lane, c0, c1);
    int col = lane & 15, rb = (lane >> 4) << 3;
    for (int r = 0; r < 8; ++r) {
      sbuf[(size_t)(i0 + rb + r) * 128 + n0 + col]      = c0[r] + bs[n0 + col];
      sbuf[(size_t)(i0 + rb + r) * 128 + n0 + 16 + col] = c1[r] + bs[n0 + 16 + col];
    }
  }
}

// Fused per-16-edge pipeline: geometry -> Fourier MLP -> r_emb -> LN -> k_e/v_e -> sim + atomic max.
__global__ __launch_bounds__(128) void k_edge_pass1(
    const int* __restrict__ ei, int E,
    const float* __restrict__ pos_a, const float* __restrict__ head_a,
    const float* __restrict__ pos_pl, const float* __restrict__ orient_pl,
    const float* __restrict__ freqs,
    const float* __restrict__ fW1, const float* __restrict__ fb1,
    const float* __restrict__ fln_g, const float* __restrict__ fln_b,
    const float* __restrict__ fW2, const float* __restrict__ fb2,
    const float* __restrict__ oln_g, const float* __restrict__ oln_b,
    const float* __restrict__ oW, const float* __restrict__ ob,
    const float* __restrict__ ln_r_g, const float* __restrict__ ln_r_b,
    const float* __restrict__ Wkr, const float* __restrict__ Wvr, const float* __restrict__ bvr,
    const float* __restrict__ kbuf, const float* __restrict__ vbuf, const float* __restrict__ qbuf,
    float* __restrict__ sim_out, float* __restrict__ ve_out, unsigned* __restrict__ menc) {
  __shared__ float sfe[16 * 132];  // Fourier features, K padded 129->132
  __shared__ float sh[16 * 128];
  __shared__ float sacc[16 * 128];
  __shared__ float red[256];
  __shared__ float srf[16 * 3];
  __shared__ int ssrc[16];
  __shared__ int sdst[16];

  const int tid = threadIdx.x, lane = tid & 31, wave = tid >> 5;
  const int e0 = blockIdx.x * 16;
  const int n0 = wave * 32;

  if (tid == 0) {
    __builtin_prefetch(fW1, 0, 1);
    __builtin_prefetch(fW2, 0, 1);
    __builtin_prefetch(oW, 0, 1);
    __builtin_prefetch(Wkr, 0, 1);
    __builtin_prefetch(Wvr, 0, 1);
  }
  if (tid < 16) {
    int e = e0 + tid;
    int ec = (e < E) ? e : (E - 1);
    int s = ei[ec], d = ei[E + ec];
    ssrc[tid] = s;
    sdst[tid] = d;
    int a = s % A_, ts = s / A_;
    float px = pos_a[((size_t)a * T_ + ts) * 2 + 0];
    float py = pos_a[((size_t)a * T_ + ts) * 2 + 1];
    float hd = head_a[a * T_ + ts];
    int pp = d % P_;
    float qx = pos_pl[pp * 2 + 0], qy = pos_pl[pp * 2 + 1];
    float o = orient_pl[pp];
    float rx = px - qx, ry = py - qy;
    float ro = fmodf(hd - o + PI_, TWO_PI_);
    if (ro < 0.f) ro += TWO_PI_;
    ro -= PI_;
    float co = cosf(o), so = sinf(o);
    float ang = atan2f(co * ry - so * rx, co * rx + so * ry);
    srf[tid * 3 + 0] = sqrtf(rx * rx + ry * ry);
    srf[tid * 3 + 1] = ang;
    srf[tid * 3 + 2] = ro;
  }
  __syncthreads();

  v8f acc0 = vzero8(), acc1 = vzero8();
  for (int ch = 0; ch < 3; ++ch) {
    for (int idx = tid; idx < 16 * 132; idx += 128) {
      int m = idx / 132, j = idx - m * 132;
      float rf = srf[m * 3 + ch];
      float v;
      if (j < 64)        v = cosf(TWO_PI_ * freqs[ch * 64 + j] * rf);
      else if (j < 128)  v = sinf(TWO_PI_ * freqs[ch * 64 + (j - 64)] * rf);
      else if (j == 128) v = rf;
      else               v = 0.f;
      sfe[idx] = v;
    }
    __syncthreads();
    v8f c0 = vzero8(), c1 = vzero8();
    gemm16(sfe, 132, fW1 + (size_t)ch * 129 * 128, 128, 132, 129, n0, lane, c0, c1);
    {
      int col = lane & 15, rb = (lane >> 4) << 3;
      for (int r = 0; r < 8; ++r) {
        sh[(rb + r) * 128 + n0 + col]      = c0[r] + fb1[ch * 128 + n0 + col];
        sh[(rb + r) * 128 + n0 + 16 + col] = c1[r] + fb1[ch * 128 + n0 + 16 + col];
      }
    }
    __syncthreads();
    ln_rows(sh, 128, 128, fln_g + ch * 128, fln_b + ch * 128, red, tid, true);
    gemm16(sh, 128, fW2 + (size_t)ch * 128 * 128, 128, 128, 128, n0, lane, acc0, acc1);
    __syncthreads();
  }
  store_c_lds(sh, 128, n0, lane, acc0, acc1);
  __syncthreads();
  for (int idx = tid; idx < 2048; idx += 128) {
    int j = idx & 127;
    sh[idx] += fb2[j] + fb2[128 + j] + fb2[256 + j];
  }
  __syncthreads();
  ln_rows(sh, 128, 128, oln_g, oln_b, red, tid, true);
  {
    v8f c0 = vzero8(), c1 = vzero8();
    gemm16(sh, 128, oW, 128, 128, 128, n0, lane, c0, c1);
    int col = lane & 15, rb = (lane >> 4) << 3;
    for (int r = 0; r < 8; ++r) {
      sacc[(rb + r) * 128 + n0 + col]      = c0[r] + ob[n0 + col];
      sacc[(rb + r) * 128 + n0 + 16 + col] = c1[r] + ob[n0 + 16 + col];
    }
  }
  __syncthreads();
  ln_rows(sacc, 128, 128, ln_r_g, ln_r_b, red, tid, false);  // rn in sacc
  {  // k_e = rn@Wkr + k[src] -> sh
    v8f c0 = vzero8(), c1 = vzero8();
    gemm16(sacc, 128, Wkr, 128, 128, 128, n0, lane, c0, c1);
    int col = lane & 15, rb = (lane >> 4) << 3;
    for (int r = 0; r < 8; ++r) {
      int m = rb + r;
      sh[m * 128 + n0 + col]      = c0[r] + kbuf[(size_t)ssrc[m] * 128 + n0 + col];
      sh[m * 128 + n0 + 16 + col] = c1[r] + kbuf[(size_t)ssrc[m] * 128 + n0 + 16 + col];
    }
  }
  {  // v_e = rn@Wvr + bvr + v[src] -> global
    v8f c0 = vzero8(), c1 = vzero8();
    gemm16(sacc, 128, Wvr, 128, 128, 128, n0, lane, c0, c1);
    int col = lane & 15, rb = (lane >> 4) << 3;
    for (int r = 0; r < 8; ++r) {
      int m = rb + r;
      if (e0 + m < E) {
        ve_out[(size_t)(e0 + m) * 128 + n0 + col] =
            c0[r] + bvr[n0 + col] + vbuf[(size_t)ssrc[m] * 128 + n0 + col];
        ve_out[(size_t)(e0 + m) * 128 + n0 + 16 + col] =
            c1[r] + bvr[n0 + 16 + col] + vbuf[(size_t)ssrc[m] * 128 + n0 + 16 + col];
      }
    }
  }
  __syncthreads();
  {  // sim[e][h] and segment max via order-preserving uint atomicMax
    int m = tid & 15, hh = tid >> 4;
    if (e0 + m < E) {
      const float* qrow = qbuf + (size_t)(sdst[m] % P_) * 128 + hh * 16;
      const float* ke = sh + m * 128 + hh * 16;
      float s = 0.f;
      for (int j = 0; j < 16; ++j) s += qrow[j] * ke[j];
      s *= 0.25f;  // HD^-0.5
      sim_out[(size_t)(e0 + m) * 8 + hh] = s;
      unsigned bb = __float_as_uint(s);
      unsigned key = (bb & 0x80000000u) ? ~bb : (bb | 0x80000000u);
      atomicMax(&menc[sdst[m] * 8 + hh], key);
    }
  }
}

__global__ void k_edge_pass2(const float* __restrict__ simb, const float* __restrict__ veb,
                             const int* __restrict__ ei, int E,
                             const unsigned* __restrict__ menc,
                             float* __restrict__ den, float* __restrict__ agg) {
  int i = blockIdx.x * blockDim.x + threadIdx.x;
  if (i >= E * 8) return;
  int e = i >> 3, h = i & 7;
  int d = ei[E + e];
  unsigned key = menc[d * 8 + h];
  unsigned bb = (key & 0x80000000u) ? (key ^ 0x80000000u) : ~key;
  float m = __uint_as_float(bb);
  float ex = expf(simb[i] - m);
  atomicAdd(&den[d * 8 + h], ex);
  const float* v = veb + (size_t)e * 128 + h * 16;
  float* ag = agg + (size_t)d * 128 + h * 16;
  for (int j = 0; j < 16; ++j) atomicAdd(&ag[j], ex * v[j]);
}

// Per-dst: normalize agg, gate (K=256 GEMM), output proj, residual, FFN 128->512->128.
__global__ __launch_bounds__(128) void k_dst_final(
    const float* __restrict__ agg, const float* __restrict__ den,
    const float* __restrict__ xdln, const float* __restrict__ sbuf,
    const float* __restrict__ map_enc,
    const float* __restrict__ Wg, const float* __restrict__ bg,
    const float* __restrict__ Wo, const float* __restrict__ bo,
    const float* __restrict__ postn_g, const float* __restrict__ postn_b,
    const float* __restrict__ ffpre_g, const float* __restrict__ ffpre_b,
    const float* __restrict__ ffW1, const float* __restrict__ ffb1,
    const float* __restrict__ ffW2, const float* __restrict__ ffb2,
    const float* __restrict__ ffpost_g, const float* __restrict__ ffpost_b,
    float* __restrict__ out) {
  __shared__ float sx[16 * 128];
  __shared__ float sA[16 * 256];
  __shared__ float sg[16 * 128];
  __shared__ float sf[16 * 256];
  __shared__ float red[256];
  const int tid = threadIdx.x, lane = tid & 31, wave = tid >> 5;
  const int d0 = blockIdx.x * 16;
  const int n0 = wave * 32;

  if (tid == 0) {
    __builtin_prefetch(Wg, 0, 1);
    __builtin_prefetch(Wo, 0, 1);
    __builtin_prefetch(ffW1, 0, 1);
    __builtin_prefetch(ffW2, 0, 1);
  }
  // xdln rows (d0..d0+15) % P are contiguous: async copy into second half of sA
  for (int idx = tid; idx < 512; idx += 128) {
    int m = idx >> 5, c = (idx & 31) << 2;
    g2l_b128(sA + m * 256 + 128 + c, xdln + (size_t)((d0 + m) % P_) * 128 + c);
  }
  for (int idx = tid; idx < 2048; idx += 128) {
    int m = idx >> 7, j = idx & 127;
    int d = d0 + m;
    sA[m * 256 + j] = agg[(size_t)d * 128 + j] / (den[d * 8 + (j >> 4)] + 1e-16f);
  }
  g2l_fence();
  __syncthreads();
  {  // g = sigmoid([agg, xd] @ Wg + bg), K = 256
    v8f c0 = vzero8(), c1 = vzero8();
    gemm16(sA, 256, Wg, 128, 256, 256, n0, lane, c0, c1);
    int col = lane & 15, rb = (lane >> 4) << 3;
    for (int r = 0; r < 8; ++r) {
      sg[(rb + r) * 128 + n0 + col]      = sigm(c0[r] + bg[n0 + col]);
      sg[(rb + r) * 128 + n0 + 16 + col] = sigm(c1[r] + bg[n0 + 16 + col]);
    }
  }
  __syncthreads();
  for (int idx = tid; idx < 2048; idx += 128) {  // o_attn = agg + g*(s - agg)
    int m = idx >> 7, j = idx & 127;
    int d = d0 + m;
    float ag = sA[m * 256 + j];
    float sv = sbuf[(size_t)(d % P_) * 128 + j];
    sx[idx] = ag + sg[idx] * (sv - ag);
  }
  __syncthreads();
  {  // y = o_attn @ Wo + bo
    v8f c0 = vzero8(), c1 = vzero8();
    gemm16(sx, 128, Wo, 128, 128, 128, n0, lane, c0, c1);
    int col = lane & 15, rb = (lane >> 4) << 3;
    for (int r = 0; r < 8; ++r) {
      sg[(rb + r) * 128 + n0 + col]      = c0[r] + bo[n0 + col];
      sg[(rb + r) * 128 + n0 + 16 + col] = c1[r] + bo[n0 + 16 + col];
    }
  }
  __syncthreads();
  ln_rows(sg, 128, 128, postn_g, postn_b, red, tid, false);
  for (int idx = tid; idx < 2048; idx += 128) {  // x = map_enc + LN(...)
    int m = idx >> 7, j = idx & 127;
    sx[idx] = map_enc[(size_t)((d0 + m) % P_) * 128 + j] + sg[idx];
  }
  __syncthreads();
  for (int idx = tid; idx < 2048; idx += 128) sg[idx] = sx[idx];
  __syncthreads();
  ln_rows(sg, 128, 128, ffpre_g, ffpre_b, red, tid, false);
  // FFN: accumulate h2 over two 256-wide halves of the hidden dim (keeps LDS < 64KB)
  v8f h20 = vzero8(), h21 = vzero8();
  for (int half = 0; half < 2; ++half) {
    for (int nb = 0; nb < 2; ++nb) {
      int ng = half * 256 + nb * 128 + n0;
      v8f c0 = vzero8(), c1 = vzero8();
      gemm16(sg, 128, ffW1, 512, 128, 128, ng, lane, c0, c1);
      int col = lane & 15, rb = (lane >> 4) << 3;
      for (int r = 0; r < 8; ++r) {
        sf[(rb + r) * 256 + nb * 128 + n0 + col]      = fmaxf(c0[r] + ffb1[ng + col], 0.f);
        sf[(rb + r) * 256 + nb * 128 + n0 + 16 + col] = fmaxf(c1[r] + ffb1[ng + 16 + col], 0.f);
      }
    }
    __syncthreads();
    gemm16(sf, 256, ffW2 + (size_t)half * 256 * 128, 128, 256, 256, n0, lane, h20, h21);
    __syncthreads();
  }
  {
    int col = lane & 15, rb = (lane >> 4) << 3;
    for (int r = 0; r < 8; ++r) {
      sg[(rb + r) * 128 + n0 + col]      = h20[r] + ffb2[n0 + col];
      sg[(rb + r) * 128 + n0 + 16 + col] = h21[r] + ffb2[n0 + 16 + col];
    }
  }
  __syncthreads();
  ln_rows(sg, 128, 128, ffpost_g, ffpost_b, red, tid, false);
  for (int idx = tid; idx < 2048; idx += 128) {
    int m = idx >> 7;
    out[(size_t)(d0 + m) * 128 + (idx & 127)] = sx[idx] + sg[idx];
  }
}

// One GRU step: each block owns 16 h-rows (safe in-place update).
__global__ __launch_bounds__(128) void k_gru(const float* __restrict__ scen, float* __restrict__ hbuf,
                                             const float* __restrict__ Wih, const float* __restrict__ Whh,
                                             const float* __restrict__ bih, const float* __restrict__ bhh,
                                             int t) {
  __shared__ float sx[16 * 128];
  __shared__ float shh[16 * 128];
  __shared__ float s1[16 * 384];
  __shared__ float shn[16 * 128];
  const int tid = threadIdx.x, lane = tid & 31, wave = tid >> 5;
  const int r0 = blockIdx.x * 16;
  const int n0 = wave * 32;

  if (tid == 0) { __builtin_prefetch(Wih, 0, 1); __builtin_prefetch(Whh, 0, 1); }
  for (int idx = tid; idx < 512; idx += 128) {
    g2l_b128(sx + idx * 4, scen + ((size_t)t * P_ + r0) * 128 + idx * 4);
    g2l_b128(shh + idx * 4, hbuf + (size_t)r0 * 128 + idx * 4);
  }
  g2l_fence();
  __syncthreads();
  for (int nb = 0; nb < 3; ++nb) {  // gi = x@Wih + bih
    int ng = nb * 128 + n0;
    v8f c0 = vzero8(), c1 = vzero8();
    gemm16(sx, 128, Wih, 384, 128, 128, ng, lane, c0, c1);
    int col = lane & 15, rb = (lane >> 4) << 3;
    for (int r = 0; r < 8; ++r) {
      s1[(rb + r) * 384 + ng + col]      = c0[r] + bih[ng + col];
      s1[(rb + r) * 384 + ng + 16 + col] = c1[r] + bih[ng + 16 + col];
    }
  }
  __syncthreads();
  for (int nb = 0; nb < 3; ++nb) {  // gh = h@Whh + bhh; r/z gates summed in place, hn kept apart
    int ng = nb * 128 + n0;
    v8f c0 = vzero8(), c1 = vzero8();
    gemm16(shh, 128, Whh, 384, 128, 128, ng, lane, c0, c1);
    int col = lane & 15, rb = (lane >> 4) << 3;
    for (int r = 0; r < 8; ++r) {
      float g0 = c0[r] + bhh[ng + col];
      float g1 = c1[r] + bhh[ng + 16 + col];
      if (ng < 256) {
        s1[(rb + r) * 384 + ng + col]      += g0;
        s1[(rb + r) * 384 + ng + 16 + col] += g1;
      } else {
        shn[(rb + r) * 128 + (ng - 256) + col]      = g0;
        shn[(rb + r) * 128 + (ng - 256) + 16 + col] = g1;
      }
    }
  }
  __syncthreads();
  for (int idx = tid; idx < 2048; idx += 128) {
    int m = idx >> 7, j = idx & 127;
    float r_ = sigm(s1[m * 384 + j]);
    float z_ = sigm(s1[m * 384 + 128 + j]);
    float n_ = tanhf(s1[m * 384 + 256 + j] + r_ * shn[m * 128 + j]);
    hbuf[(size_t)(r0 + m) * 128 + j] = (1.f - z_) * n_ + z_ * shh[idx];
  }
}

__global__ void k_copy_h(const float* __restrict__ hbuf, float* __restrict__ out) {
  int i = blockIdx.x * blockDim.x + threadIdx.x;
  if (i < P_ * HID_) out[i] = hbuf[i];
}

// ---------------------------------------------------------------------------
extern "C" void kernel_launch(void* const* d_in, const int* in_sizes, int n_in,
                              void* d_out, int out_size, void* d_ws, size_t ws_size,
                              hipStream_t stream) {
  (void)n_in; (void)out_size; (void)ws_size;
  const float* pos_a     = (const float*)d_in[0];
  const float* head_a    = (const float*)d_in[1];
  const float* pos_pl    = (const float*)d_in[2];
  const float* orient_pl = (const float*)d_in[3];
  const float* map_enc   = (const float*)d_in[4];
  const float* agent_enc = (const float*)d_in[5];
  // params dict, insertion order
  const float* freqs    = (const float*)d_in[6];
  const float* fW1      = (const float*)d_in[7];
  const float* fb1      = (const float*)d_in[8];
  const float* fln_g    = (const float*)d_in[9];
  const float* fln_b    = (const float*)d_in[10];
  const float* fW2      = (const float*)d_in[11];
  const float* fb2      = (const float*)d_in[12];
  const float* oln_g    = (const float*)d_in[13];
  const float* oln_b    = (const float*)d_in[14];
  const float* oW       = (const float*)d_in[15];
  const float* ob       = (const float*)d_in[16];
  const float* ln_src_g = (const float*)d_in[17];
  const float* ln_src_b = (const float*)d_in[18];
  const float* ln_dst_g = (const float*)d_in[19];
  const float* ln_dst_b = (const float*)d_in[20];
  const float* ln_r_g   = (const float*)d_in[21];
  const float* ln_r_b   = (const float*)d_in[22];
  const float* Wq       = (const float*)d_in[23];
  const float* bq       = (const float*)d_in[24];
  const float* Wk       = (const float*)d_in[25];
  const float* Wv       = (const float*)d_in[26];
  const float* bv       = (const float*)d_in[27];
  const float* Wkr      = (const float*)d_in[28];
  const float* Wvr      = (const float*)d_in[29];
  const float* bvr      = (const float*)d_in[30];
  const float* Ws       = (const float*)d_in[31];
  const float* bs       = (const float*)d_in[32];
  const float* Wg       = (const float*)d_in[33];
  const float* bg       = (const float*)d_in[34];
  const float* Wo       = (const float*)d_in[35];
  const float* bo       = (const float*)d_in[36];
  const float* postn_g  = (const float*)d_in[37];
  const float* postn_b  = (const float*)d_in[38];
  const float* ffpre_g  = (const float*)d_in[39];
  const float* ffpre_b  = (const float*)d_in[40];
  const float* ffW1     = (const float*)d_in[41];
  const float* ffb1     = (const float*)d_in[42];
  const float* ffW2     = (const float*)d_in[43];
  const float* ffb2     = (const float*)d_in[44];
  const float* ffpost_g = (const float*)d_in[45];
  const float* ffpost_b = (const float*)d_in[46];
  const float* Wih      = (const float*)d_in[47];
  const float* Whh      = (const float*)d_in[48];
  const float* bih      = (const float*)d_in[49];
  const float* bhh      = (const float*)d_in[50];
  const float* gru_h0   = (const float*)d_in[51];
  const int*   ei       = (const int*)d_in[52];
  const int E = in_sizes[52] / 2;

  float* ws = (float*)d_ws;
  size_t off = 0;
  unsigned* menc = (unsigned*)(ws + off); off += (size_t)NDST_ * 8;
  float* den  = ws + off; off += (size_t)NDST_ * 8;
  float* agg  = ws + off; off += (size_t)NDST_ * 128;
  float* kbuf = ws + off; off += (size_t)NSRC_ * 128;
  float* vbuf = ws + off; off += (size_t)NSRC_ * 128;
  float* qbuf = ws + off; off += (size_t)P_ * 128;
  float* sbuf = ws + off; off += (size_t)P_ * 128;
  float* xdln = ws + off; off += (size_t)P_ * 128;
  float* hbuf = ws + off; off += (size_t)P_ * 128;
  float* simb = ws + off; off += (size_t)E * 8;
  float* veb  = ws + off; off += (size_t)E * 128;

  float* out_scen = (float*)d_out;
  float* out_h = out_scen + (size_t)T_ * P_ * HID_;

  k_init<<<(NDST_ * 128 + 255) / 256, 256, 0, stream>>>(menc, den, agg, hbuf, gru_h0);
  k_src<<<NSRC_ / 16, 128, 0, stream>>>(agent_enc, ln_src_g, ln_src_b, Wk, Wv, bv, kbuf, vbuf);
  k_dst<<<P_ / 16, 128, 0, stream>>>(map_enc, ln_dst_g, ln_dst_b, Wq, bq, Ws, bs, qbuf, sbuf, xdln);
  if (E > 0) {
    k_edge_pass1<<<(E + 15) / 16, 128, 0, stream>>>(
        ei, E, pos_a, head_a, pos_pl, orient_pl, freqs, fW1, fb1, fln_g, fln_b, fW2, fb2,
        oln_g, oln_b, oW, ob, ln_r_g, ln_r_b, Wkr, Wvr, bvr, kbuf, vbuf, qbuf,
        simb, veb, menc);
    k_edge_pass2<<<(E * 8 + 255) / 256, 256, 0, stream>>>(simb, veb, ei, E, menc, den, agg);
  }
  k_dst_final<<<NDST_ / 16, 128, 0, stream>>>(agg, den, xdln, sbuf, map_enc, Wg, bg, Wo, bo,
                                              postn_g, postn_b, ffpre_g, ffpre_b, ffW1, ffb1,
                                              ffW2, ffb2, ffpost_g, ffpost_b, out_scen);
  for (int t = 0; t < T_; ++t)
    k_gru<<<P_ / 16, 128, 0, stream>>>(out_scen, hbuf, Wih, Whh, bih, bhh, t);
  k_copy_h<<<(P_ * 128 + 255) / 256, 256, 0, stream>>>(hbuf, out_h);
}